// MixOp_1494648619468
// MI455X (gfx1250) — compile-verified
//
#include <hip/hip_runtime.h>

typedef float v2f __attribute__((ext_vector_type(2)));
typedef float v8f __attribute__((ext_vector_type(8)));

#define EPSBN 1e-5f

// ---------------------------------------------------------------------------
// Kernel 1: on-device routing + weight/BN preparation (1 wave32).
// ws layout:  int  wsI[0]=index, wsI[1]=K (1/3/5), wsI[2]=R (=K/2)
//             float wsF[0..3]=bn scale, wsF[4..7]=bn bias,
//             wsF[8..407] = 25 tap matrices, each 4x4 (ci-major, co-minor)
// Dense HWIO (Kh,Kw,4,4) flat layout == tap-matrix flat layout, so the dense
// fill is a straight copy; depthwise scatters onto the diagonal.
// ---------------------------------------------------------------------------
__global__ void route_setup(const float* __restrict__ logits,
                            const float* __restrict__ g,
                            const float* __restrict__ w0,
                            const float* __restrict__ w1,
                            const float* __restrict__ w2,
                            const float* __restrict__ w3,
                            const float* __restrict__ w4,
                            const float* __restrict__ gamma,
                            const float* __restrict__ beta,
                            const float* __restrict__ mean,
                            const float* __restrict__ var,
                            int* __restrict__ wsI,
                            float* __restrict__ wsF) {
  __shared__ int sidx;
  const int lane = threadIdx.x;
  if (lane == 0) {
    // argmax((logits+g)/TAU) == argmax(logits+g), TAU>0 uniform
    float best = logits[0] + g[0];
    int idx = 0;
    for (int i = 1; i < 5; ++i) {
      float v = logits[i] + g[i];
      if (v > best) { best = v; idx = i; }
    }
    const int K = (idx == 0) ? 1 : ((idx < 3) ? 3 : 5);
    wsI[0] = idx;
    wsI[1] = K;
    wsI[2] = K >> 1;
    // fused inference BN; final hard scale is exactly 1.0 (onehot+soft-soft)
    for (int c = 0; c < 4; ++c) {
      float inv = gamma[idx * 4 + c] * rsqrtf(var[idx * 4 + c] + EPSBN);
      wsF[c]     = inv;
      wsF[4 + c] = beta[idx * 4 + c] - mean[idx * 4 + c] * inv;
    }
    sidx = idx;
  }
  __syncthreads();

  const int idx = sidx;
  const int K  = (idx == 0) ? 1 : ((idx < 3) ? 3 : 5);
  const int KK = K * K;
  const float* wsel = (idx == 0) ? w0 : (idx == 1) ? w1
                    : (idx == 2) ? w2 : (idx == 3) ? w3 : w4;
  const bool dw = (idx == 2) || (idx == 4);
  float* taps = wsF + 8;
  for (int i = lane; i < 25 * 16; i += 32) {
    const int t  = i >> 4;
    const int e  = i & 15;
    const int ci = e >> 2;
    const int co = e & 3;
    float v = 0.f;
    if (t < KK) {
      if (!dw)            v = wsel[i];          // dense: identical flat layout
      else if (ci == co)  v = wsel[t * 4 + co]; // depthwise: diagonal
    }
    taps[i] = v;
  }
}

// ---------------------------------------------------------------------------
// Kernel 2: selected-branch conv + BN + ReLU via V_WMMA_F32_16X16X4_F32,
// with the input halo patch DMA'd into LDS via GLOBAL_LOAD_ASYNC_TO_LDS_B128.
// Block: 256 threads = 8 wave32; block tile = 128 pixels of one output row.
// ---------------------------------------------------------------------------
__global__ __launch_bounds__(256) void branch_conv_wmma(
    const float* __restrict__ x,  // (32,512,512,4) NHWC fp32
    const int* __restrict__ wsI,
    const float* __restrict__ wsF,
    float* __restrict__ out) {
  __shared__ float4 patch[5 * 132];   // up to 5 rows x (128+4) halo cols x 4ch
  __shared__ float  tapsS[25 * 16];   // tap matrices
  __shared__ float  sb[8];            // bn scale[4], bias[4]
  __shared__ float  stage[8 * 64];    // per-wave output transpose staging

  const int tid = threadIdx.x;
  const int K = wsI[1];               // kernel width (1/3/5), block-uniform
  const int R = wsI[2];               // radius

  const int blk   = blockIdx.x;       // 65536 blocks: 4 tiles * 512 h * 32 n
  const int wtile = blk & 3;
  const int h     = (blk >> 2) & 511;
  const int n     = blk >> 11;
  const int wbase = wtile * 128;

  if (tid < 25 * 16) tapsS[tid] = wsF[8 + tid];
  if (tid < 8) sb[tid] = wsF[tid];

  // ---- async-DMA halo patch into LDS (zero padded via ds stores) ----
  const int cols = 128 + 2 * R;
  const int rows = K;
  for (int i = tid; i < rows * cols; i += 256) {
    const int r  = i / cols;
    const int c  = i - r * cols;
    const int gh = h - R + r;
    const int gw = wbase - R + c;
    if (gh >= 0 && gh < 512 && gw >= 0 && gw < 512) {
      const float* gp = x + (size_t)((n * 512 + gh) * 512 + gw) * 4;
      const unsigned ldsOff = (unsigned)(unsigned long long)
          (__attribute__((address_space(3))) char*)(&patch[r * 132 + c]);
      const unsigned long long ga = (unsigned long long)gp;
      // GV mode: vdst = per-lane LDS byte address, vaddr = 64-bit global addr
      asm volatile("global_load_async_to_lds_b128 %0, %1, off"
                   :: "v"(ldsOff), "v"(ga) : "memory");
    } else {
      patch[r * 132 + c] = make_float4(0.f, 0.f, 0.f, 0.f);
    }
  }
  // each wave drains its own tensor DMA before signaling the block barrier
  asm volatile("s_wait_asynccnt 0x0" ::: "memory");
  __syncthreads();

  // ---- per-wave WMMA accumulation over taps ----
  const int wave = tid >> 5;
  const int lane = tid & 31;
  const int m    = lane & 15;   // A row (pixel) / B column (co) index
  const int half = lane >> 4;   // selects K-pair {0,1} vs {2,3} (ISA layout)
  const int wsub = wave * 16;

  v8f acc = {0.f, 0.f, 0.f, 0.f, 0.f, 0.f, 0.f, 0.f};
  for (int kh = 0; kh < K; ++kh) {
    const float4* prow = &patch[kh * 132 + wsub + m];
    for (int kw = 0; kw < K; ++kw) {
      const float4 p = prow[kw];
      v2f a;                                   // A[m][k], k = 2*half+{0,1}
      a.x = half ? p.z : p.x;
      a.y = half ? p.w : p.y;
      const int t = kh * K + kw;
      v2f b;                                   // B[k][n], n = m (cols>=4 -> 0)
      b.x = (m < 4) ? tapsS[t * 16 + (half * 2 + 0) * 4 + m] : 0.f;
      b.y = (m < 4) ? tapsS[t * 16 + (half * 2 + 1) * 4 + m] : 0.f;
      acc = __builtin_amdgcn_wmma_f32_16x16x4_f32(
          /*neg_a=*/false, a, /*neg_b=*/false, b,
          /*c_mod=*/(short)0, acc, /*reuse_a=*/false, /*reuse_b=*/false);
    }
  }

  // ---- BN + ReLU on the 4 used D columns, stage pixel-major in LDS ----
  if (m < 4) {
    const float s  = sb[m];
    const float bb = sb[4 + m];
    const int mbase = half * 8;       // D rows 0-7 in lanes 0-15, 8-15 in 16-31
#pragma unroll
    for (int r = 0; r < 8; ++r) {
      float v = acc[r] * s + bb;
      v = v > 0.f ? v : 0.f;
      stage[wave * 64 + (mbase + r) * 4 + m] = v;
    }
  }
  __syncthreads();

  // ---- coalesced store: 64 floats per wave = one float2 per lane ----
  const size_t pixBase = (size_t)((n * 512 + h) * 512 + wbase + wave * 16);
  float2 v2;
  v2.x = stage[wave * 64 + lane * 2 + 0];
  v2.y = stage[wave * 64 + lane * 2 + 1];
  reinterpret_cast<float2*>(out + pixBase * 4)[lane] = v2;
}

// ---------------------------------------------------------------------------
extern "C" void kernel_launch(void* const* d_in, const int* in_sizes, int n_in,
                              void* d_out, int out_size, void* d_ws,
                              size_t ws_size, hipStream_t stream) {
  (void)in_sizes; (void)n_in; (void)out_size; (void)ws_size;
  const float* x      = (const float*)d_in[0];
  const float* logits = (const float*)d_in[1];
  const float* g      = (const float*)d_in[2];
  const float* w0     = (const float*)d_in[3];
  const float* w1     = (const float*)d_in[4];
  const float* w2     = (const float*)d_in[5];
  const float* w3     = (const float*)d_in[6];
  const float* w4     = (const float*)d_in[7];
  const float* gamma  = (const float*)d_in[8];
  const float* beta   = (const float*)d_in[9];
  const float* mean   = (const float*)d_in[10];
  const float* var    = (const float*)d_in[11];

  int*   wsI = (int*)d_ws;
  float* wsF = (float*)d_ws + 16;  // 64-byte offset past the int header

  route_setup<<<1, 32, 0, stream>>>(logits, g, w0, w1, w2, w3, w4,
                                    gamma, beta, mean, var, wsI, wsF);

  // 32 batch * 512 rows * 4 tiles-of-128 = 65536 blocks
  branch_conv_wmma<<<dim3(65536), dim3(256), 0, stream>>>(
      x, wsI, wsF, (float*)d_out);
}